// NodeModel_20203526160533
// MI455X (gfx1250) — compile-verified
//
#include <hip/hip_runtime.h>

typedef float v2f __attribute__((ext_vector_type(2)));
typedef float v8f __attribute__((ext_vector_type(8)));

#define N_NODES 100000
#define E_EDGES 640000
#define V_IN_D  128
#define U_IN_D  64
#define HID_D   128
#define LDS_STRIDE 132   // pad 128->132 floats: conflict-free LDS access
#define WAVES_PER_BLOCK 2
#define M_TILE 32        // rows per wave (two 16-row WMMA fragments)

// ---------------- kernel 1: zero the agg scratch ----------------
__global__ void zero_kernel(float* __restrict__ p, int n) {
    int i = blockIdx.x * blockDim.x + threadIdx.x;
    if (i < n) p[i] = 0.0f;
}

// ---------------- kernel 2: scatter-add edge_attr into agg ----------------
// one thread per (edge, 4-float chunk): 32 threads (1 wave) cover one edge row of 128
__global__ void scatter_kernel(const float* __restrict__ edge_attr,
                               const int* __restrict__ col,
                               float* __restrict__ agg) {
    long idx = (long)blockIdx.x * blockDim.x + threadIdx.x;
    int e = (int)(idx >> 5);
    int f = ((int)idx & 31) * 4;
    if (e >= E_EDGES) return;
    int dst = col[e];
    const float* src = edge_attr + (size_t)e * HID_D + f;
    float* d = agg + (size_t)dst * HID_D + f;
#pragma unroll
    for (int j = 0; j < 4; ++j)
        unsafeAtomicAdd(d + j, src[j]);   // global_atomic_add_f32
}

// ---------------- fused MLP with f32 WMMA ----------------
__device__ __forceinline__ v8f wmma4(v2f a, v2f b, v8f c) {
    return __builtin_amdgcn_wmma_f32_16x16x4_f32(
        false, a, false, b, (short)0, c, false, false);
}

// one K-step: two A frags (32 rows) vs 8 shared B-frag column tiles
__device__ __forceinline__ void wmma_step2(v2f a0, v2f a1,
                                           const float* __restrict__ w0,
                                           v8f acc0[8], v8f acc1[8]) {
#pragma unroll
    for (int t = 0; t < 8; ++t) {
        v2f b;
        b.x = w0[t * 16];           // W[k0+2*hi    ][t*16 + lo]
        b.y = w0[HID_D + t * 16];   // W[k0+2*hi + 1][t*16 + lo]
        acc0[t] = wmma4(a0, b, acc0[t]);
        acc1[t] = wmma4(a1, b, acc1[t]);   // reuse B fragment
    }
}

__global__ __launch_bounds__(WAVES_PER_BLOCK * 32)
void mlp_kernel(const float* __restrict__ x,
                const float* __restrict__ agg,
                const float* __restrict__ u,
                const int* __restrict__ batch,
                const float* __restrict__ W1,
                const float* __restrict__ b1,
                const float* __restrict__ W2,
                const float* __restrict__ b2,
                float* __restrict__ out) {
    __shared__ float h1s[WAVES_PER_BLOCK * M_TILE * LDS_STRIDE];

    const int wave = threadIdx.x >> 5;
    const int lane = threadIdx.x & 31;
    const int lo = lane & 15;
    const int hi = lane >> 4;          // 0 or 1

    const long mtile = (long)blockIdx.x * WAVES_PER_BLOCK + wave;
    const long m0 = mtile * M_TILE;
    if (m0 >= N_NODES) return;         // uniform per-wave exit; N % 32 == 0

    const long row0 = m0 + lo;         // rows of A-frag 0
    const long row1 = m0 + 16 + lo;    // rows of A-frag 1
    const float* xrow0 = x + (size_t)row0 * V_IN_D;
    const float* xrow1 = x + (size_t)row1 * V_IN_D;
    const float* aggrow0 = agg + (size_t)row0 * HID_D;
    const float* aggrow1 = agg + (size_t)row1 * HID_D;
    const float* urow0 = u + (size_t)batch[row0] * U_IN_D;
    const float* urow1 = u + (size_t)batch[row1] * U_IN_D;

    v8f zero8 = {};
    v8f acc0[8], acc1[8];
#pragma unroll
    for (int t = 0; t < 8; ++t) { acc0[t] = zero8; acc1[t] = zero8; }

    // ---- GEMM1: concat([x, agg, u[batch]]) @ W1, K = 320 in 3 sub-loops ----
    for (int k0 = 0; k0 < V_IN_D; k0 += 4) {
        v2f a0 = *(const v2f*)(xrow0 + k0 + 2 * hi);
        v2f a1 = *(const v2f*)(xrow1 + k0 + 2 * hi);
        const float* w0 = W1 + (size_t)(k0 + 2 * hi) * HID_D + lo;
        wmma_step2(a0, a1, w0, acc0, acc1);
    }
    for (int k0 = 0; k0 < HID_D; k0 += 4) {
        v2f a0 = *(const v2f*)(aggrow0 + k0 + 2 * hi);
        v2f a1 = *(const v2f*)(aggrow1 + k0 + 2 * hi);
        const float* w0 = W1 + (size_t)(V_IN_D + k0 + 2 * hi) * HID_D + lo;
        wmma_step2(a0, a1, w0, acc0, acc1);
    }
    for (int k0 = 0; k0 < U_IN_D; k0 += 4) {
        v2f a0 = *(const v2f*)(urow0 + k0 + 2 * hi);
        v2f a1 = *(const v2f*)(urow1 + k0 + 2 * hi);
        const float* w0 = W1 + (size_t)(V_IN_D + HID_D + k0 + 2 * hi) * HID_D + lo;
        wmma_step2(a0, a1, w0, acc0, acc1);
    }

    // ---- epilogue 1: bias + ReLU, stash h1 tile (32 x 128) in LDS ----
    float* hbase = h1s + wave * M_TILE * LDS_STRIDE;
#pragma unroll
    for (int t = 0; t < 8; ++t) {
        float bv = b1[t * 16 + lo];
#pragma unroll
        for (int r = 0; r < 8; ++r) {
            float v0 = fmaxf(acc0[t][r] + bv, 0.0f);
            float v1 = fmaxf(acc1[t][r] + bv, 0.0f);
            hbase[(r + 8 * hi) * LDS_STRIDE + t * 16 + lo] = v0;
            hbase[(16 + r + 8 * hi) * LDS_STRIDE + t * 16 + lo] = v1;
        }
        acc0[t] = zero8;               // reuse accumulators for GEMM2
        acc1[t] = zero8;
    }

    // ---- GEMM2: h1 @ W2, K = 128, A-frags re-loaded from LDS ----
    for (int k0 = 0; k0 < HID_D; k0 += 4) {
        v2f a0 = *(const v2f*)(hbase + lo * LDS_STRIDE + k0 + 2 * hi);
        v2f a1 = *(const v2f*)(hbase + (16 + lo) * LDS_STRIDE + k0 + 2 * hi);
        const float* w0 = W2 + (size_t)(k0 + 2 * hi) * HID_D + lo;
        wmma_step2(a0, a1, w0, acc0, acc1);
    }

    // ---- epilogue 2: bias + store ----
#pragma unroll
    for (int t = 0; t < 8; ++t) {
        float bv = b2[t * 16 + lo];
#pragma unroll
        for (int r = 0; r < 8; ++r) {
            out[(size_t)(m0 + r + 8 * hi) * HID_D + t * 16 + lo] = acc0[t][r] + bv;
            out[(size_t)(m0 + 16 + r + 8 * hi) * HID_D + t * 16 + lo] = acc1[t][r] + bv;
        }
    }
}

extern "C" void kernel_launch(void* const* d_in, const int* in_sizes, int n_in,
                              void* d_out, int out_size, void* d_ws, size_t ws_size,
                              hipStream_t stream) {
    const float* x         = (const float*)d_in[0];
    const int*   edgeidx   = (const int*)d_in[1];   // (2, E)
    const float* edge_attr = (const float*)d_in[2];
    const float* u         = (const float*)d_in[3];
    const int*   batch     = (const int*)d_in[4];
    const float* W1        = (const float*)d_in[5];
    const float* b1        = (const float*)d_in[6];
    const float* W2        = (const float*)d_in[7];
    const float* b2        = (const float*)d_in[8];
    float* out = (float*)d_out;

    float* agg = (float*)d_ws;                      // N * HID floats = 51.2 MB
    const int* col = edgeidx + E_EDGES;             // edge_index[1]

    // 1) zero agg
    int zn = N_NODES * HID_D;
    zero_kernel<<<(zn + 511) / 512, 512, 0, stream>>>(agg, zn);

    // 2) scatter-add edge features
    long st = (long)E_EDGES * 32;
    scatter_kernel<<<(unsigned)((st + 255) / 256), 256, 0, stream>>>(edge_attr, col, agg);

    // 3) fused concat + MLP (WMMA f32), 32 rows per wave
    int tiles = (N_NODES + M_TILE - 1) / M_TILE;                // 3125
    int blocks = (tiles + WAVES_PER_BLOCK - 1) / WAVES_PER_BLOCK;
    mlp_kernel<<<blocks, WAVES_PER_BLOCK * 32, 0, stream>>>(
        x, agg, u, batch, W1, b1, W2, b2, out);
}